// CustomCrossEntropyLoss_18562848654175
// MI455X (gfx1250) — compile-verified
//
#include <hip/hip_runtime.h>

// CDNA5 / gfx1250: wave32, WMMA 16x16x4 f32 used for the block-level reduction.
typedef __attribute__((ext_vector_type(2))) float v2f;
typedef __attribute__((ext_vector_type(8))) float v8f;

#define HW_SHIFT 18           // H*W = 512*512 = 2^18
#define NUM_CLASSES 4
#define IGNORE_INDEX 4
#define THREADS 256           // 8 wave32s per block

// Per-pixel masked weighted NLL accumulation (stable log-softmax over 4 classes).
__device__ __forceinline__ void px_accum(float a, float b, float c, float d,
                                         int t, int me, float& loss, float& cnt) {
  float m = fmaxf(fmaxf(a, b), fmaxf(c, d));
  float s = __expf(a - m) + __expf(b - m) + __expf(c - m) + __expf(d - m);
  float lse = m + __logf(s);
  float xt = a;
  xt = (t == 1) ? b : xt;
  xt = (t == 2) ? c : xt;
  xt = (t == 3) ? d : xt;
  if (t != IGNORE_INDEX) {
    float w = (me == 0) ? 1.0f : 0.5f;
    loss += w * (lse - xt);
    cnt  += 1.0f;
  }
}

__global__ __launch_bounds__(THREADS)
void ce_partial_kernel(const float* __restrict__ pred,
                       const int*  __restrict__ tgt,
                       const int*  __restrict__ me,
                       float2*     __restrict__ part,
                       int ngroups) {
  __shared__ float lds_loss[THREADS];
  __shared__ float lds_cnt[THREADS];

  const int gid = blockIdx.x * THREADS + threadIdx.x;
  float loss = 0.0f, cnt = 0.0f;

  if (gid < ngroups) {
    const int p0 = gid << 2;                       // first of 4 consecutive pixels
    const int b  = p0 >> HW_SHIFT;                 // batch index (HW divisible by 4)
    const int hw = p0 & ((1 << HW_SHIFT) - 1);
    const size_t base = ((size_t)b << (HW_SHIFT + 2)) + (size_t)hw; // b*C*HW + hw

    // 6 x b128 loads per thread: 4 class streams + target + ME (all coalesced).
    const float4 x0 = *(const float4*)(pred + base);
    const float4 x1 = *(const float4*)(pred + base + (size_t)(1 << HW_SHIFT));
    const float4 x2 = *(const float4*)(pred + base + (size_t)(2 << HW_SHIFT));
    const float4 x3 = *(const float4*)(pred + base + (size_t)(3 << HW_SHIFT));
    const int4  t4 = *(const int4*)(tgt + p0);
    const int4  m4 = *(const int4*)(me  + p0);

    px_accum(x0.x, x1.x, x2.x, x3.x, t4.x, m4.x, loss, cnt);
    px_accum(x0.y, x1.y, x2.y, x3.y, t4.y, m4.y, loss, cnt);
    px_accum(x0.z, x1.z, x2.z, x3.z, t4.z, m4.z, loss, cnt);
    px_accum(x0.w, x1.w, x2.w, x3.w, t4.w, m4.w, loss, cnt);
  }

  lds_loss[threadIdx.x] = loss;
  lds_cnt [threadIdx.x] = cnt;
  __syncthreads();

  // Wave 0 reduces the 256-element tiles with chained V_WMMA_F32_16X16X4_F32.
  // A = ones(16x4)  =>  D[m][n] = sum_k B[k][n]  (column sums, all D rows equal).
  // 4 chained WMMAs cover a full 16x16 f32 tile; every staged value lands in
  // exactly one column, so summing D row 0 over lanes 0..15 yields the total.
  if (threadIdx.x < 32) {                          // whole wave32 -> EXEC all ones
    const int lane = threadIdx.x;
    v2f aones; aones[0] = 1.0f; aones[1] = 1.0f;   // A[m][k] = 1 for all slots
    v8f accL = {};
    v8f accC = {};
#pragma unroll
    for (int j = 0; j < 4; ++j) {
      v2f bl, bc;
      bl[0] = lds_loss[j * 64 + lane];
      bl[1] = lds_loss[j * 64 + 32 + lane];
      bc[0] = lds_cnt [j * 64 + lane];
      bc[1] = lds_cnt [j * 64 + 32 + lane];
      accL = __builtin_amdgcn_wmma_f32_16x16x4_f32(false, aones, false, bl,
                                                   (short)0, accL, false, false);
      accC = __builtin_amdgcn_wmma_f32_16x16x4_f32(false, aones, false, bc,
                                                   (short)0, accC, false, false);
    }
    // D VGPR0: lanes 0-15 -> M=0,N=lane; lanes 16-31 -> M=8,N=lane-16.
    // Sum the 16 column sums within each 16-lane half (masks 1,2,4,8).
    float l = accL[0];
    float c = accC[0];
#pragma unroll
    for (int msk = 1; msk <= 8; msk <<= 1) {
      l += __shfl_xor(l, msk, 32);
      c += __shfl_xor(c, msk, 32);
    }
    if (lane == 0) part[blockIdx.x] = make_float2(l, c);
  }
}

__global__ __launch_bounds__(THREADS)
void ce_final_kernel(const float2* __restrict__ part, int nparts,
                     float* __restrict__ out) {
  __shared__ float lds_l[8];
  __shared__ float lds_c[8];
  float l = 0.0f, c = 0.0f;
  for (int i = threadIdx.x; i < nparts; i += THREADS) {
    float2 p = part[i];
    l += p.x;
    c += p.y;
  }
#pragma unroll
  for (int msk = 16; msk >= 1; msk >>= 1) {
    l += __shfl_xor(l, msk, 32);
    c += __shfl_xor(c, msk, 32);
  }
  const int wave = threadIdx.x >> 5;
  const int lane = threadIdx.x & 31;
  if (lane == 0) { lds_l[wave] = l; lds_c[wave] = c; }
  __syncthreads();
  if (threadIdx.x == 0) {
    float L = 0.0f, C = 0.0f;
#pragma unroll
    for (int w = 0; w < 8; ++w) { L += lds_l[w]; C += lds_c[w]; }
    out[0] = L / C;
  }
}

extern "C" void kernel_launch(void* const* d_in, const int* in_sizes, int n_in,
                              void* d_out, int out_size, void* d_ws, size_t ws_size,
                              hipStream_t stream) {
  const float* pred = (const float*)d_in[0];   // [16,4,512,512] f32
  const int*   tgt  = (const int*)d_in[1];     // [16,512,512] i32
  const int*   me   = (const int*)d_in[2];     // [16,512,512] i32
  float*       out  = (float*)d_out;           // scalar f32

  const int npix    = in_sizes[1];             // B*H*W = 4,194,304
  const int ngroups = npix >> 2;               // 4 pixels per thread
  const int nblocks = (ngroups + THREADS - 1) / THREADS;   // 4096

  float2* part = (float2*)d_ws;                // nblocks * 8 bytes of scratch

  ce_partial_kernel<<<nblocks, THREADS, 0, stream>>>(pred, tgt, me, part, ngroups);
  ce_final_kernel<<<1, THREADS, 0, stream>>>(part, nblocks, out);
}